// OFT_45028437131813
// MI455X (gfx1250) — compile-verified
//
#include <hip/hip_runtime.h>
#include <hip/hip_bf16.h>
#include <math.h>

// ---------------------------------------------------------------------------
// Problem constants (B=1)
// ---------------------------------------------------------------------------
#define NCAM 6
#define CCH  64
#define HF   64
#define WF   176
#define ZD   129
#define XD   129
#define NY   4
#define NH   3
#define DD   128
#define WD   128
#define WROW (WF * CCH)          // 11264, row stride of channel-last integral image
#define CAMSTRIDE (HF * WROW)    // 720896 floats per camera

typedef __attribute__((ext_vector_type(2))) float v2f;
typedef __attribute__((ext_vector_type(8))) float v8f;

// ---------------------------------------------------------------------------
// Kernel 1: project all grid corners -> normalized clipped coords nc[n][iy][iz][ix][2]
// ---------------------------------------------------------------------------
__global__ void OFT_proj_kernel(const float* __restrict__ ks,
                                const float* __restrict__ imu2cs,
                                const float* __restrict__ post_rots,
                                const float* __restrict__ post_trans,
                                const float* __restrict__ undists,
                                const float* __restrict__ grid,
                                float* __restrict__ nc) {
  int t = blockIdx.x * blockDim.x + threadIdx.x;
  const int total = NCAM * NY * ZD * XD;
  if (t >= total) return;
  int ix = t % XD;
  int r  = t / XD;
  int iz = r % ZD;
  r /= ZD;
  int iy = r % NY;
  int n  = r / NY;

  const float* K  = ks        + n * 9;
  const float* M  = imu2cs    + n * 12;
  const float* PR = post_rots + n * 9;
  const float* PT = post_trans + n * 3;
  const float* Du = undists   + n * 7;

  // calib = K(3x3) @ imu2c(3x4)
  float cal[12];
  #pragma unroll
  for (int i = 0; i < 3; ++i)
    #pragma unroll
    for (int j = 0; j < 4; ++j)
      cal[i * 4 + j] = K[i * 3 + 0] * M[0 * 4 + j] +
                       K[i * 3 + 1] * M[1 * 4 + j] +
                       K[i * 3 + 2] * M[2 * 4 + j];

  const float* g = grid + (iz * XD + ix) * 3;
  float px3 = g[0];
  float py3 = g[1] + (2.0f - (float)iy);   // ys = [2,1,0,-1]
  float pz3 = g[2];

  float hx = cal[0] * px3 + cal[1] * py3 + cal[2]  * pz3 + cal[3];
  float hy = cal[4] * px3 + cal[5] * py3 + cal[6]  * pz3 + cal[7];
  float hz = cal[8] * px3 + cal[9] * py3 + cal[10] * pz3 + cal[11];

  float fl = (hz > 0.0f) ? 1.0f : 0.0f;    // behind-camera flag
  float px = (hx * fl) / hz;
  float py = (hy * fl) / hz;

  float fx = K[0], fy = K[4], ccx = K[2], ccy = K[5];
  float x = (px - ccx) / fx;
  float y = (py - ccy) / fy;

  float ptsx, ptsy;
  if (Du[6] == 1.0f) {
    // fisheye
    float rr = sqrtf(x * x + y * y);
    float th = atanf(rr);
    float t2 = th * th;
    float rad = th * (1.0f + t2 * (Du[0] + t2 * (Du[1] + t2 * (Du[2] + t2 * Du[5])))) / rr;
    ptsx = (x * rad * fx + ccx) * fl;
    ptsy = (y * rad * fy + ccy) * fl;
  } else {
    // pinhole with tangential terms
    float r2   = x * x + y * y;
    float poly = 1.0f + r2 * (Du[0] + r2 * (Du[1] + r2 * Du[2]));
    float xd = x * poly + (2.0f * Du[3] * x * y + Du[4] * (r2 + 2.0f * x * x));
    float yd = y * poly + (Du[3] * (r2 + 2.0f * y * y) + 2.0f * Du[4] * x * y);
    ptsx = (xd * fx + ccx) * fl;
    ptsy = (yd * fy + ccy) * fl;
  }

  float qx = PR[0] * ptsx + PR[1] * ptsy + PT[0];
  float qy = PR[3] * ptsx + PR[4] * ptsy + PT[1];
  float ncx = fminf(fmaxf(2.0f * qx - 1.0f, -1.0f), 1.0f);
  float ncy = fminf(fmaxf(2.0f * qy - 1.0f, -1.0f), 1.0f);

  nc[t * 2 + 0] = ncx;
  nc[t * 2 + 1] = ncy;
}

// ---------------------------------------------------------------------------
// Kernel 2: row prefix-sum + transpose to channel-last: T1[n][y][x][c]
// block = (n,y): 64 threads = channels; stores are 256B-coalesced across c.
// ---------------------------------------------------------------------------
__global__ void OFT_scan_x_kernel(const float* __restrict__ feat,
                                  float* __restrict__ T1) {
  int n = blockIdx.x / HF;
  int y = blockIdx.x % HF;
  int c = threadIdx.x;
  const float* src = feat + ((size_t)(n * CCH + c) * HF + y) * WF;
  float* dst = T1 + (size_t)(n * HF + y) * WROW + c;
  float s = 0.0f;
  for (int x = 0; x < WF; ++x) {
    s += src[x];
    dst[x * CCH] = s;
  }
}

// ---------------------------------------------------------------------------
// Kernel 3: column cumsum as triangular-ones matmul via V_WMMA_F32_16X16X4_F32.
//   S[y, xc] = sum_{y'<=y} T1[y', xc]   (M = y tiles of 16, K = y' tiles of 4,
//   N = xc tiles of 16; 704 N-tiles per camera).
// A (16x4, documented layout): lane m=lane&15; VGPR j holds K = 2*(lane>=16)+j.
// B (4x16, per §7.12.2 striping): lanes 0-15 hold K=0,1 across the 2 VGPRs,
//   lanes 16-31 hold K=2,3; N = lane&15.
// D (16x16 f32, documented): VGPR r -> M = r + 8*(lane>=16), N = lane&15.
// EXEC is all ones (no divergence) as WMMA requires.
// ---------------------------------------------------------------------------
__global__ void OFT_scan_y_wmma_kernel(const float* __restrict__ T1,
                                       float* __restrict__ S) {
  int b = blockIdx.x;
  int nchunk = b % 88;  b /= 88;        // 704 ntiles / 8 waves per block
  int mtile  = b % 4;
  int n      = b / 4;
  int wave = threadIdx.x >> 5;
  int lane = threadIdx.x & 31;
  int ntile = nchunk * 8 + wave;

  int nloc = lane & 15;                 // N index (and A's M row index)
  int hi   = lane >> 4;                 // selects K pair {0,1} vs {2,3}
  int k0   = hi * 2;

  const float* src = T1 + (size_t)n * CAMSTRIDE + ntile * 16 + nloc;
  float*       dst = S  + (size_t)n * CAMSTRIDE + ntile * 16 + nloc;

  v8f acc = {0.f, 0.f, 0.f, 0.f, 0.f, 0.f, 0.f, 0.f};
  int mrow = mtile * 16 + nloc;         // global output row handled by A row m
  int numK = 4 * mtile + 4;             // triangular: only K-tiles up to diagonal

  for (int kt = 0; kt < numK; ++kt) {
    int kb = kt * 4;
    v2f a, bm;
    // lower-triangular ones: L[mrow, kb+k] = (kb+k <= mrow)
    a.x = (kb + k0 + 0 <= mrow) ? 1.0f : 0.0f;
    a.y = (kb + k0 + 1 <= mrow) ? 1.0f : 0.0f;
    bm.x = src[(size_t)(kb + k0 + 0) * WROW];
    bm.y = src[(size_t)(kb + k0 + 1) * WROW];
    acc = __builtin_amdgcn_wmma_f32_16x16x4_f32(false, a, false, bm,
                                                (short)0, acc, false, false);
  }

  #pragma unroll
  for (int r = 0; r < 8; ++r) {
    dst[(size_t)(mtile * 16 + r + hi * 8) * WROW] = acc[r];
  }
}

// ---------------------------------------------------------------------------
// Kernel 4: per-voxel box-average from integral images + max over cameras.
// block = one voxel (iy,iz,ix), thread = channel c. All taps are 256B bursts.
// ---------------------------------------------------------------------------
__device__ __forceinline__ void OFT_prep(float cc, float m, int& i0, int& i1, float& w) {
  // grid_sample align_corners=True, border padding; m = (dim-1) as float
  float u = (cc + 1.0f) * 0.5f * m;
  u = fminf(fmaxf(u, 0.0f), m);
  float f = floorf(u);
  i0 = (int)f;
  w  = u - f;
  i1 = min(i0 + 1, (int)m);
}

__device__ __forceinline__ float OFT_bil(const float* __restrict__ base, int c,
                                         int y0, int y1, float wy,
                                         int x0, int x1, float wx) {
  const float* r0 = base + (size_t)y0 * WROW;
  const float* r1 = base + (size_t)y1 * WROW;
  float v00 = r0[x0 * CCH + c];
  float v01 = r0[x1 * CCH + c];
  float v10 = r1[x0 * CCH + c];
  float v11 = r1[x1 * CCH + c];
  return v00 * (1.0f - wx) * (1.0f - wy) + v01 * wx * (1.0f - wy) +
         v10 * (1.0f - wx) * wy          + v11 * wx * wy;
}

__global__ void OFT_sample_kernel(const float* __restrict__ nc,
                                  const float* __restrict__ integ,
                                  float* __restrict__ out) {
  int b = blockIdx.x;                   // iy*16384 + iz*128 + ix
  int c = threadIdx.x;                  // 0..63
  int ix = b % WD;
  int r  = b / WD;
  int iz = r % DD;
  int iy = r / DD;

  float acc = -INFINITY;
  #pragma unroll 1
  for (int n = 0; n < NCAM; ++n) {
    // bbox of the voxel's 8 projected corners (uniform across the block)
    const float* ncb = nc + (size_t)(((n * NY + iy) * ZD + iz) * XD + ix) * 2;
    float l = 1e30f, rt = -1e30f, tp = 1e30f, bt = -1e30f;
    #pragma unroll
    for (int dy = 0; dy < 2; ++dy)
      #pragma unroll
      for (int dz = 0; dz < 2; ++dz)
        #pragma unroll
        for (int dx = 0; dx < 2; ++dx) {
          const float* p = ncb + (size_t)((dy * ZD + dz) * XD + dx) * 2;
          float vx = p[0], vy = p[1];
          l  = fminf(l, vx);  rt = fmaxf(rt, vx);
          tp = fminf(tp, vy); bt = fmaxf(bt, vy);
        }

    float area = (rt - l) * (bt - tp) * (float)(HF * WF) * 0.25f + 1e-6f;
    float w = (area > 1e-6f) ? (1.0f / area) : 0.0f;

    int xl0, xl1, xr0, xr1, yt0, yt1, yb0, yb1;
    float wxl, wxr, wyt, wyb;
    OFT_prep(l,  (float)(WF - 1), xl0, xl1, wxl);
    OFT_prep(rt, (float)(WF - 1), xr0, xr1, wxr);
    OFT_prep(tp, (float)(HF - 1), yt0, yt1, wyt);
    OFT_prep(bt, (float)(HF - 1), yb0, yb1, wyb);

    const float* base = integ + (size_t)n * CAMSTRIDE;
    float tl = OFT_bil(base, c, yt0, yt1, wyt, xl0, xl1, wxl);
    float tr = OFT_bil(base, c, yt0, yt1, wyt, xr0, xr1, wxr);
    float bl = OFT_bil(base, c, yb0, yb1, wyb, xl0, xl1, wxl);
    float br = OFT_bil(base, c, yb0, yb1, wyb, xr0, xr1, wxr);

    float val = (tl + br - tr - bl) * w;
    acc = fmaxf(acc, val);
  }

  out[(size_t)((c * NH + iy) * DD + iz) * WD + ix] = acc;
}

// ---------------------------------------------------------------------------
// Launcher
// ---------------------------------------------------------------------------
extern "C" void kernel_launch(void* const* d_in, const int* in_sizes, int n_in,
                              void* d_out, int out_size, void* d_ws, size_t ws_size,
                              hipStream_t stream) {
  (void)in_sizes; (void)n_in; (void)out_size; (void)ws_size;
  const float* features  = (const float*)d_in[0];
  const float* ks        = (const float*)d_in[1];
  const float* imu2cs    = (const float*)d_in[2];
  const float* post_rots = (const float*)d_in[3];
  const float* post_trans= (const float*)d_in[4];
  const float* undists   = (const float*)d_in[5];
  const float* grid      = (const float*)d_in[6];
  float* out = (float*)d_out;

  float* ws = (float*)d_ws;
  const size_t SZ_NC = (size_t)NCAM * NY * ZD * XD * 2;   //   798,768 f
  const size_t SZ_T1 = (size_t)NCAM * CAMSTRIDE;          // 4,325,376 f
  float* nc = ws;
  float* T1 = ws + SZ_NC;
  float* S  = ws + SZ_NC + SZ_T1;

  const int total_pts = NCAM * NY * ZD * XD;
  OFT_proj_kernel<<<(total_pts + 255) / 256, 256, 0, stream>>>(
      ks, imu2cs, post_rots, post_trans, undists, grid, nc);

  OFT_scan_x_kernel<<<NCAM * HF, CCH, 0, stream>>>(features, T1);

  // 6 cams * 4 mtiles * 88 ntile-chunks, 8 waves/block (one ntile per wave)
  OFT_scan_y_wmma_kernel<<<NCAM * 4 * 88, 256, 0, stream>>>(T1, S);

  OFT_sample_kernel<<<NH * DD * WD, CCH, 0, stream>>>(nc, S, out);
}